// SphericalBessel_44203803411087
// MI455X (gfx1250) — compile-verified
//
#include <hip/hip_runtime.h>
#include <cstdint>
#include <cstddef>

// SphericalBessel: N=65536 radii, K=32, L=32, downward Miller recurrence in f64.
// One wave32 per radius n; lane = k-1. Output tile out[n, 0:32, 0:32] is 4 KB,
// staged in LDS with identical layout, then streamed out with the CDNA5 async
// LDS->global store engine (global_store_async_from_lds_b128, ASYNCcnt),
// with non-temporal hint since the 256 MB output is write-once.

#define K_MAX  32
#define L_MAX  32
#define LSTART 49            // L_MAX + int(sqrt(10*L_MAX)) = 32 + 17
#define WAVES_PER_BLOCK 8

__global__ __launch_bounds__(256)
void SphericalBessel_kernel(const float* __restrict__ r,
                            float* __restrict__ out,
                            int n_total) {
    const int lane = threadIdx.x & 31;
    const int wave = threadIdx.x >> 5;
    const int n    = blockIdx.x * WAVES_PER_BLOCK + wave;

    __shared__ float smem[WAVES_PER_BLOCK * L_MAX * K_MAX];   // 32 KB

    if (n >= n_total) return;   // wave-uniform: whole wave exits together

    const int    k    = lane + 1;
    const double x    = (double)r[n] * (double)k;
    const double invx = 1.0 / x;

    // true j0(x) = sin(x)/x -- computed first to keep its live range off the
    // 32-double result array.
    const double sinc = sin(x) * invx;

    // Downward Miller recurrence, fully unrolled, f64 (iterates reach ~1e127).
    double j1 = 1.0, j2 = 0.0;
#pragma unroll
    for (int i = LSTART; i > L_MAX; --i) {
        double j0 = (double)(2 * i + 1) * invx * j1 - j2;
        j2 = j1; j1 = j0;
    }
    double y[L_MAX];
#pragma unroll
    for (int i = L_MAX; i >= 1; --i) {
        double j0 = (double)(2 * i + 1) * invx * j1 - j2;
        y[i - 1] = j0;
        j2 = j1; j1 = j0;
    }

    // Normalize Miller iterates and fold in sqrt(2/pi)*k.
    const double scale = (sinc / y[0]) * (0.79788456080286535587989211986876 * (double)k);

    // Stage f32 tile in LDS with the exact global layout: [l][k].
    float* wbase = &smem[wave * (L_MAX * K_MAX)];
#pragma unroll
    for (int l = 0; l < L_MAX; ++l) {
        wbase[l * K_MAX + lane] = (float)(y[l] * scale);   // bank-conflict free
    }

    // Drain this wave's ds_stores before the async engine reads LDS.
    asm volatile("s_wait_dscnt 0x0" ::: "memory");

    // 4 KB tile: LDS layout == global layout -> 8 rounds of per-lane 16 B
    // async LDS->global stores (512 B per wave per round). Fire-and-forget,
    // tracked by ASYNCcnt, no VGPR round-trip. NT hint: output is streaming,
    // write-once (256 MB vs 192 MB L2) -- don't pollute cache.
    const uint32_t lds_base = (uint32_t)(uintptr_t)wbase;  // low 32 bits of a
                                                           // generic LDS ptr ==
                                                           // raw LDS byte offset
    const uint64_t gbase =
        (uint64_t)(uintptr_t)(out + (size_t)n * (L_MAX * K_MAX));
#pragma unroll
    for (int t = 0; t < 8; ++t) {
        const uint32_t off   = (uint32_t)(t * 512 + lane * 16);
        const uint64_t gaddr = gbase + off;
        const uint32_t laddr = lds_base + off;
        asm volatile("global_store_async_from_lds_b128 %0, %1, off th:TH_STORE_NT"
                     :: "v"(gaddr), "v"(laddr)
                     : "memory");
    }
    // Make completion explicit (S_ENDPGM also implies wait-idle).
    asm volatile("s_wait_asynccnt 0x0" ::: "memory");
}

extern "C" void kernel_launch(void* const* d_in, const int* in_sizes, int n_in,
                              void* d_out, int out_size, void* d_ws, size_t ws_size,
                              hipStream_t stream) {
    const float* r   = (const float*)d_in[0];
    float*       out = (float*)d_out;
    const int n_total = in_sizes[0];                     // 65536
    const int blocks  = (n_total + WAVES_PER_BLOCK - 1) / WAVES_PER_BLOCK;
    SphericalBessel_kernel<<<blocks, 256, 0, stream>>>(r, out, n_total);
}